// TransTransducer_63376537420404
// MI455X (gfx1250) — compile-verified
//
#include <hip/hip_runtime.h>
#include <hip/hip_bf16.h>
#include <math.h>

// ---------------- model constants ----------------
#define BB   4
#define TT   256
#define UU   64
#define UU1  65
#define HH   144
#define CC   320
#define LL   16
#define NHH  8
#define DHH  18
#define DFFN 2048
#define VV   128
#define EPSL 1e-5f

typedef __attribute__((ext_vector_type(2))) float v2f;
typedef __attribute__((ext_vector_type(8))) float v8f;

// f32 WMMA 16x16x4: D = A(16x4) * B(4x16) + C, wave32.
// A frag: lane l in [0,16): row l; vgpr pair = K {2*half, 2*half+1}
// B frag: lane l = column;   vgpr pair = K {2*half, 2*half+1}
// C/D:    row = j + 8*half, col = lane l
__device__ __forceinline__ v8f wmma4(v2f a, v2f b, v8f c) {
#if defined(__HIP_DEVICE_COMPILE__) && __has_builtin(__builtin_amdgcn_wmma_f32_16x16x4_f32)
  return __builtin_amdgcn_wmma_f32_16x16x4_f32(false, a, false, b, (short)0, c, false, false);
#else
  (void)a; (void)b; return c;   // host pass / missing-builtin fallback
#endif
}

// ---------------- embedding gather ----------------
__global__ __launch_bounds__(256) void embed_kernel(const int* __restrict__ xs,
                                                    const float* __restrict__ emb,
                                                    float* __restrict__ x) {
  int i = blockIdx.x * 256 + threadIdx.x;
  if (i < BB * TT * HH) {
    int m = i / HH, h = i - m * HH;
    x[i] = emb[xs[m] * HH + h];
  }
}

// ---------------- generic GEMM: Out[m,n] = act(sum_k A[m,k]*W[n,k] + bias[n]) ----------------
// A: M x K row-major (lda = K, K even). W rows have stride ldW (even). N multiple of 16.
__global__ __launch_bounds__(256) void gemm_wmma(const float* __restrict__ A,
                                                 const float* __restrict__ W,
                                                 const float* __restrict__ bias,
                                                 float* __restrict__ Out,
                                                 int M, int N, int K, int ldW, int act) {
  __shared__ float As[16][36];
  int tid  = threadIdx.x;
  int w    = tid >> 5;
  int lane = tid & 31;
  int l    = lane & 15;
  int half = lane >> 4;
  int m0 = blockIdx.y * 16;
  int n0 = blockIdx.x * 128 + w * 16;
  bool valid = (n0 < N);                 // wave-uniform
  bool mfull = (m0 + 16 <= M);           // block-uniform
  // staging coords: each thread moves one float2 of the 16x32 A tile
  int sr = tid >> 4;                     // 0..15
  int sc = (tid & 15) * 2;               // 0..30 even
  const float* Arow;
  {
    int gm = m0 + sr;
    int gms = (gm < M) ? gm : (M - 1);
    Arow = A + (size_t)gms * K;
  }
  bool min_ = (m0 + sr) < M;
  const float* Wrow = W + (size_t)(n0 + l) * ldW + 2 * half;  // per-lane row base
  v8f acc = {};

  int Kfull = K & ~31;                   // full 32-wide chunks (no k bounds checks)
  int k0 = 0;
  for (; k0 < Kfull; k0 += 32) {
    if (mfull) {
      *(v2f*)&As[sr][sc] = *(const v2f*)(Arow + k0 + sc);        // b64 -> b64, no guards
    } else {
      v2f av = *(const v2f*)(Arow + k0 + sc);                    // clamped row
      av.x = min_ ? av.x : 0.f;
      av.y = min_ ? av.y : 0.f;
      *(v2f*)&As[sr][sc] = av;
    }
    __syncthreads();
    if (valid) {
      if (k0 + 32 < K) __builtin_prefetch(Wrow + k0 + 32, 0, 1);
      #pragma unroll
      for (int kk = 0; kk < 32; kk += 4) {
        int kp = kk + 2 * half;
        v2f a = *(const v2f*)&As[l][kp];          // ds_load_b64
        v2f b = *(const v2f*)(Wrow + k0 + kk);    // global_load_b64
        acc = wmma4(a, b, acc);
      }
    }
    __syncthreads();
  }

  if (k0 < K) {                          // single guarded remainder chunk (K=144 case)
    {
      int gk = k0 + sc;
      bool kin = (gk < K);               // K even => gk+1 < K iff gk < K
      int gks = kin ? gk : (K - 2);
      v2f av = *(const v2f*)(Arow + gks);
      bool in = min_ && kin;
      av.x = in ? av.x : 0.f;
      av.y = in ? av.y : 0.f;
      *(v2f*)&As[sr][sc] = av;
    }
    __syncthreads();
    if (valid) {
      #pragma unroll
      for (int kk = 0; kk < 32; kk += 4) {
        int gk = k0 + kk + 2 * half;
        v2f a = *(const v2f*)&As[l][kk + 2 * half];
        bool kin = (gk < K);
        int gs = kin ? gk : (K - 2);
        v2f braw = *(const v2f*)(W + (size_t)(n0 + l) * ldW + gs);
        v2f b;
        b.x = kin ? braw.x : 0.f;
        b.y = kin ? braw.y : 0.f;
        acc = wmma4(a, b, acc);
      }
    }
    __syncthreads();
  }

  if (valid) {
    float bv = bias ? bias[n0 + l] : 0.f;
    if (mfull) {
      #pragma unroll
      for (int jj = 0; jj < 8; ++jj) {
        float v = acc[jj] + bv;
        if (act == 1) v = fmaxf(v, 0.f);
        Out[(m0 + jj + 8 * half) * N + n0 + l] = v;
      }
    } else {
      #pragma unroll
      for (int jj = 0; jj < 8; ++jj) {
        int row = m0 + jj + 8 * half;
        if (row < M) {
          float v = acc[jj] + bv;
          if (act == 1) v = fmaxf(v, 0.f);
          Out[row * N + n0 + l] = v;
        }
      }
    }
  }
}

// ---------------- attention: per (b,head) block, per-thread online softmax ----------------
__global__ __launch_bounds__(256) void attn_kernel(const float* __restrict__ qkv,
                                                   float* __restrict__ obuf) {
  int bh = blockIdx.x;
  int b = bh >> 3, h = bh & 7;
  __shared__ float ks[TT][DHH];
  __shared__ float vs[TT][DHH];
  int tid = threadIdx.x;
  for (int i = tid; i < TT * DHH; i += 256) {
    int r = i / DHH, d = i - r * DHH;
    const float* base = qkv + (b * TT + r) * (3 * HH);
    ks[r][d] = base[HH     + h * DHH + d];
    vs[r][d] = base[2 * HH + h * DHH + d];
  }
  __syncthreads();
  int t = tid;
  float q[DHH], acc[DHH];
  const float* qb = qkv + (b * TT + t) * (3 * HH) + h * DHH;
  #pragma unroll
  for (int d = 0; d < DHH; ++d) { q[d] = qb[d]; acc[d] = 0.f; }
  const float scale = 0.2357022603955158f;  // 1/sqrt(18)
  float mrun = -INFINITY, lrun = 0.f;
  for (int s = 0; s < TT; ++s) {
    float dot = 0.f;
    #pragma unroll
    for (int d = 0; d < DHH; ++d) dot += q[d] * ks[s][d];
    float sc = dot * scale;
    float nm = fmaxf(mrun, sc);
    float corr = expf(mrun - nm);
    float p = expf(sc - nm);
    lrun = lrun * corr + p;
    #pragma unroll
    for (int d = 0; d < DHH; ++d) acc[d] = acc[d] * corr + p * vs[s][d];
    mrun = nm;
  }
  float invl = 1.f / lrun;
  float* ob = obuf + (b * TT + t) * HH + h * DHH;
  #pragma unroll
  for (int d = 0; d < DHH; ++d) ob[d] = acc[d] * invl;
}

// ---------------- residual add + layernorm (in-place on x) ----------------
__global__ __launch_bounds__(256) void addln_kernel(float* __restrict__ x,
                                                    const float* __restrict__ t,
                                                    const float* __restrict__ g,
                                                    const float* __restrict__ bb) {
  int row = blockIdx.x * 256 + threadIdx.x;   // 1024 rows
  const float* xr = x + row * HH;
  const float* tr = t + row * HH;
  float s = 0.f, ss = 0.f;
  for (int i = 0; i < HH; ++i) { float v = xr[i] + tr[i]; s += v; ss += v * v; }
  float m = s * (1.f / HH);
  float var = ss * (1.f / HH) - m * m;
  float inv = 1.f / sqrtf(var + EPSL);
  float* xw = x + row * HH;
  for (int i = 0; i < HH; ++i) { float v = xr[i] + tr[i]; xw[i] = (v - m) * inv * g[i] + bb[i]; }
}

// ---------------- LSTM decoder: one block per batch, h in LDS ----------------
__global__ __launch_bounds__(320) void lstm_kernel(const int* __restrict__ ys,
                                                   const float* __restrict__ wih,
                                                   const float* __restrict__ whh,
                                                   const float* __restrict__ bvec,
                                                   float* __restrict__ gout) {
  int b = blockIdx.x;
  int cidx = threadIdx.x;   // 0..319
  __shared__ float h[CC];
  float c0 = 0.f;
  h[cidx] = 0.f;
  __syncthreads();
  for (int u = 0; u < UU1; ++u) {
    int idx = (u == 0) ? -1 : (ys[b * UU + (u - 1)] - 1);  // one-hot column
    float gacc[4];
    #pragma unroll
    for (int q = 0; q < 4; ++q) {
      int j = q * CC + cidx;
      float acc = bvec[j] + ((idx >= 0) ? wih[j * (VV - 1) + idx] : 0.f);
      const float* wr = whh + j * CC;
      for (int kk = 0; kk < CC; ++kk) acc += wr[kk] * h[kk];
      gacc[q] = acc;
    }
    float ig = 1.f / (1.f + expf(-gacc[0]));
    float fg = 1.f / (1.f + expf(-gacc[1]));
    float gg = tanhf(gacc[2]);
    float og = 1.f / (1.f + expf(-gacc[3]));
    c0 = fg * c0 + ig * gg;
    float hn = og * tanhf(c0);
    __syncthreads();
    h[cidx] = hn;
    gout[(b * UU1 + u) * CC + cidx] = hn;
    __syncthreads();
  }
}

// ---------------- fused joint: tanh -> WMMA logits -> log-softmax -> lpb/lpy ----------------
__global__ __launch_bounds__(256) void joint_kernel(const float* __restrict__ fproj,
                                                    const float* __restrict__ gproj,
                                                    const float* __restrict__ fc1b,
                                                    const float* __restrict__ fc2w,
                                                    const float* __restrict__ fc2b,
                                                    const int* __restrict__ ys,
                                                    float* __restrict__ lpb,
                                                    float* __restrict__ lpy) {
  __shared__ float ht[16][322];      // tanh activations (16 rows x K=320)
  __shared__ float lg[16][132];      // logits tile (16 x 128)
  __shared__ int foffs[16], goffs[16], obi[16], oyi[16], yv[16];
  int tid = threadIdx.x;
  int m0 = blockIdx.x * 16;          // rows m = (b,t,u) flattened; M = 66560 exact
  if (tid < 16) {
    int m = m0 + tid;
    int b = m / (TT * UU1);
    int rem = m - b * (TT * UU1);
    int t = rem / UU1;
    int u = rem - t * UU1;
    foffs[tid] = (b * TT + t) * CC;
    goffs[tid] = (b * UU1 + u) * CC;
    obi[tid]   = (b * TT + t) * UU1 + u;
    oyi[tid]   = (u < UU) ? ((b * TT + t) * UU + u) : -1;
    yv[tid]    = (u < UU) ? ys[b * UU + u] : 0;
  }
  __syncthreads();
  for (int i = tid; i < 16 * CC; i += 256) {
    int r = i / CC, c = i - r * CC;
    ht[r][c] = tanhf(fproj[foffs[r] + c] + gproj[goffs[r] + c] + fc1b[c]);
  }
  __syncthreads();
  int w = tid >> 5, lane = tid & 31, l = lane & 15, half = lane >> 4;
  int n0 = w * 16;                    // 8 waves x 16 = 128 vocab cols
  const float* Wrow = fc2w + (n0 + l) * CC + 2 * half;
  v8f acc = {};
  #pragma unroll 8
  for (int k = 0; k < CC; k += 4) {
    v2f a = *(const v2f*)&ht[l][k + 2 * half];  // ds_load_b64
    v2f b = *(const v2f*)(Wrow + k);            // global_load_b64
    acc = wmma4(a, b, acc);
  }
  float bv = fc2b[n0 + l];
  #pragma unroll
  for (int jj = 0; jj < 8; ++jj) lg[jj + 8 * half][n0 + l] = acc[jj] + bv;
  __syncthreads();
  if (tid < 16) {
    int r = tid;
    float mx = -INFINITY;
    for (int v = 0; v < VV; ++v) mx = fmaxf(mx, lg[r][v]);
    float sum = 0.f;
    for (int v = 0; v < VV; ++v) sum += expf(lg[r][v] - mx);
    float lse = mx + logf(sum);
    lpb[obi[r]] = lg[r][0] - lse;
    if (oyi[r] >= 0) lpy[oyi[r]] = lg[r][yv[r]] - lse;
  }
}

// ---------------- RNNT loss: parallel scans over u, sequential over t ----------------
__device__ __forceinline__ float laef(float a, float b) {
  float m = fmaxf(a, b);
  if (m == -INFINITY) return -INFINITY;
  return m + logf(expf(a - m) + expf(b - m));
}

__global__ __launch_bounds__(512) void rnnt_kernel(const float* __restrict__ lpb,
                                                   const float* __restrict__ lpy,
                                                   float* __restrict__ out) {
  __shared__ float alpha[BB][72];
  __shared__ float s[BB][72];
  __shared__ float pf[BB][72];
  __shared__ float llsh[BB];
  int tid = threadIdx.x;
  int b = tid >> 7;       // 4 batches x 128 threads
  int j = tid & 127;

  // alpha0 = exclusive cumsum of lpy[b,0,:]
  if (j < UU) s[b][j] = lpy[(b * TT) * UU + j];
  __syncthreads();
  for (int off = 1; off < UU; off <<= 1) {
    float v = 0.f, ww = 0.f;
    if (j < UU) { v = s[b][j]; if (j >= off) ww = s[b][j - off]; }
    __syncthreads();
    if (j < UU) s[b][j] = v + ww;
    __syncthreads();
  }
  if (j < UU1) alpha[b][j] = (j == 0) ? 0.f : s[b][j - 1];
  __syncthreads();

  for (int t = 1; t < TT; ++t) {
    // pf = exclusive cumsum of lpy[b,t,:]
    if (j < UU) s[b][j] = lpy[(b * TT + t) * UU + j];
    __syncthreads();
    for (int off = 1; off < UU; off <<= 1) {
      float v = 0.f, ww = 0.f;
      if (j < UU) { v = s[b][j]; if (j >= off) ww = s[b][j - off]; }
      __syncthreads();
      if (j < UU) s[b][j] = v + ww;
      __syncthreads();
    }
    if (j < UU1) pf[b][j] = (j == 0) ? 0.f : s[b][j - 1];
    __syncthreads();
    // d = alpha + lpb[b,t-1,:] - pf ; inclusive cumlogsumexp over 65
    if (j < UU1) s[b][j] = alpha[b][j] + lpb[(b * TT + (t - 1)) * UU1 + j] - pf[b][j];
    __syncthreads();
    for (int off = 1; off < UU1; off <<= 1) {
      float v = -INFINITY, ww = -INFINITY;
      if (j < UU1) { v = s[b][j]; if (j >= off) ww = s[b][j - off]; }
      __syncthreads();
      if (j < UU1) s[b][j] = laef(v, ww);
      __syncthreads();
    }
    if (j < UU1) alpha[b][j] = pf[b][j] + s[b][j];
    __syncthreads();
  }
  if (j == 0) llsh[b] = alpha[b][UU] + lpb[(b * TT + (TT - 1)) * UU1 + UU];
  __syncthreads();
  if (tid == 0) out[0] = -0.25f * (llsh[0] + llsh[1] + llsh[2] + llsh[3]);
}

// ---------------- host orchestration ----------------
extern "C" void kernel_launch(void* const* d_in, const int* in_sizes, int n_in,
                              void* d_out, int out_size, void* d_ws, size_t ws_size,
                              hipStream_t stream) {
  (void)in_sizes; (void)n_in; (void)out_size; (void)ws_size;
  const int*   xs       = (const int*)d_in[0];
  const int*   ys       = (const int*)d_in[1];
  const float* emb_in   = (const float*)d_in[4];
  const float* enc_wqkv = (const float*)d_in[5];
  const float* enc_bqkv = (const float*)d_in[6];
  const float* enc_wo   = (const float*)d_in[7];
  const float* enc_bo   = (const float*)d_in[8];
  const float* enc_w1   = (const float*)d_in[9];
  const float* enc_b1   = (const float*)d_in[10];
  const float* enc_w2   = (const float*)d_in[11];
  const float* enc_b2   = (const float*)d_in[12];
  const float* ln1_g    = (const float*)d_in[13];
  const float* ln1_b    = (const float*)d_in[14];
  const float* ln2_g    = (const float*)d_in[15];
  const float* ln2_b    = (const float*)d_in[16];
  const float* fc0_w    = (const float*)d_in[17];
  const float* fc0_b    = (const float*)d_in[18];
  const float* lstm_wih = (const float*)d_in[19];
  const float* lstm_whh = (const float*)d_in[20];
  const float* lstm_b   = (const float*)d_in[21];
  const float* fc1_w    = (const float*)d_in[22];
  const float* fc1_b    = (const float*)d_in[23];
  const float* fc2_w    = (const float*)d_in[24];
  const float* fc2_b    = (const float*)d_in[25];

  // workspace layout (floats)
  float* ws    = (float*)d_ws;
  float* x     = ws;                     // 1024*144
  float* qkv   = x     + 147456;         // 1024*432
  float* obuf  = qkv   + 442368;         // 1024*144
  float* tmp   = obuf  + 147456;         // 1024*144
  float* ffbuf = tmp   + 147456;         // 1024*2048
  float* f     = ffbuf + 2097152;        // 1024*320
  float* g     = f     + 327680;         // 260*320
  float* fproj = g     + 83200;          // 1024*320
  float* gproj = fproj + 327680;         // 260*320
  float* lpbuf = gproj + 83200;          // 4*256*65
  float* lpybf = lpbuf + 66560;          // 4*256*64

  auto gemm = [&](const float* A, const float* W, const float* bias, float* Out,
                  int M, int N, int K, int ldW, int act) {
    dim3 grid((N + 127) / 128, (M + 15) / 16);
    gemm_wmma<<<grid, 256, 0, stream>>>(A, W, bias, Out, M, N, K, ldW, act);
  };

  embed_kernel<<<(BB * TT * HH + 255) / 256, 256, 0, stream>>>(xs, emb_in, x);

  for (int l = 0; l < LL; ++l) {
    const float* wqkv = enc_wqkv + (size_t)l * 3 * HH * HH;
    gemm(x, wqkv, enc_bqkv + l * 3 * HH, qkv, BB * TT, 3 * HH, HH, HH, 0);
    attn_kernel<<<BB * NHH, 256, 0, stream>>>(qkv, obuf);
    gemm(obuf, enc_wo + (size_t)l * HH * HH, enc_bo + l * HH, tmp, BB * TT, HH, HH, HH, 0);
    addln_kernel<<<4, 256, 0, stream>>>(x, tmp, ln1_g + l * HH, ln1_b + l * HH);
    gemm(x, enc_w1 + (size_t)l * DFFN * HH, enc_b1 + l * DFFN, ffbuf, BB * TT, DFFN, HH, HH, 1);
    gemm(ffbuf, enc_w2 + (size_t)l * HH * DFFN, enc_b2 + l * HH, tmp, BB * TT, HH, DFFN, DFFN, 0);
    addln_kernel<<<4, 256, 0, stream>>>(x, tmp, ln2_g + l * HH, ln2_b + l * HH);
  }

  gemm(x, fc0_w, fc0_b, f, BB * TT, CC, HH, HH, 0);
  lstm_kernel<<<BB, CC, 0, stream>>>(ys, lstm_wih, lstm_whh, lstm_b, g);

  // fproj = f @ wf.T (wf = fc1_w[:, :C], row stride 640); gproj = g @ wg.T (wg = fc1_w[:, C:])
  gemm(f, fc1_w,      nullptr, fproj, BB * TT,  CC, CC, 2 * CC, 0);
  gemm(g, fc1_w + CC, nullptr, gproj, BB * UU1, CC, CC, 2 * CC, 0);

  joint_kernel<<<(BB * TT * UU1) / 16, 256, 0, stream>>>(fproj, gproj, fc1_b, fc2_w, fc2_b,
                                                         ys, lpbuf, lpybf);
  rnnt_kernel<<<1, 512, 0, stream>>>(lpbuf, lpybf, (float*)d_out);
}